// GCN_63075889709536
// MI455X (gfx1250) — compile-verified
//
#include <hip/hip_runtime.h>
#include <hip/hip_bf16.h>

typedef __attribute__((ext_vector_type(2))) float v2f;
typedef __attribute__((ext_vector_type(8))) float v8f;

#define FIN  128
#define HID  64
#define NCLS 16

// ---------------- degree / normalization ----------------

__global__ void k_init_deg(float* __restrict__ deg, int n) {
    int i = blockIdx.x * blockDim.x + threadIdx.x;
    if (i < n) deg[i] = 1.0f;  // self-loop weight
}

__global__ void k_deg_accum(const long long* __restrict__ ei,
                            const float* __restrict__ ew,
                            float* __restrict__ deg, int E) {
    int e = blockIdx.x * blockDim.x + threadIdx.x;
    if (e < E) atomicAdd(&deg[(int)ei[(size_t)E + e]], ew[e]);
}

__global__ void k_dinv(float* __restrict__ deg, int n) {
    int i = blockIdx.x * blockDim.x + threadIdx.x;
    if (i < n) {
        float d = deg[i];
        deg[i] = (d > 0.0f) ? rsqrtf(d) : 0.0f;
    }
}

// ---------------- fp32 WMMA GEMM: C[M,NOUT] = (relu?)A[M,K] @ B[K,NOUT] ----------------
// One wave32 per 16x16 output tile, V_WMMA_F32_16X16X4_F32 over K in steps of 4.
// A-frag (16x4 f32): lanes 0-15 hold K=0,1 ; lanes 16-31 hold K=2,3 (row M = lane%16).
// B-frag (4x16 f32): lanes 0-15 hold K=0,1 ; lanes 16-31 hold K=2,3 (col N = lane%16).
// C/D (16x16 f32, 8 VGPRs): VGPR v -> row v (lanes 0-15) / row v+8 (lanes 16-31), col = lane%16.
template <int K, int NOUT, bool RELU>
__global__ void gemm_wmma_f32(const float* __restrict__ A,
                              const float* __restrict__ B,
                              float* __restrict__ C, int M) {
    const int ntn  = NOUT / 16;
    int wave = (int)((blockIdx.x * blockDim.x + threadIdx.x) >> 5);
    int lane = threadIdx.x & 31;
    int mtiles = (M + 15) >> 4;
    int mt = wave / ntn;
    int nt = wave - mt * ntn;
    if (mt >= mtiles) return;              // wave-uniform: EXEC stays all-1s for WMMA

    int m0 = mt << 4, n0 = nt << 4;
    int half = lane >> 4;                  // 0: K pair {0,1}, 1: K pair {2,3}
    int l = lane & 15;
    int row = m0 + l; if (row >= M) row = M - 1;   // clamp (tail store is guarded)
    const float* arow = A + (size_t)row * K + 2 * half;
    const float* bcol = B + (size_t)(2 * half) * NOUT + n0 + l;

    v8f c = {};
#pragma unroll 4
    for (int k = 0; k < K; k += 4) {
        float a0 = arow[k];
        float a1 = arow[k + 1];
        if (RELU) { a0 = fmaxf(a0, 0.0f); a1 = fmaxf(a1, 0.0f); }
        v2f a; a.x = a0; a.y = a1;
        const float* bp = bcol + (size_t)k * NOUT;
        v2f b; b.x = bp[0]; b.y = bp[NOUT];
        c = __builtin_amdgcn_wmma_f32_16x16x4_f32(
                /*neg_a=*/false, a, /*neg_b=*/false, b,
                /*c_mod=*/(short)0, c, /*reuse_a=*/false, /*reuse_b=*/false);
    }

    float* cout = C + (size_t)(m0 + half * 8) * NOUT + n0 + l;
    if (m0 + 16 <= M) {
        // full tile: unguarded, fully coalesced stores (no exec-mask churn)
#pragma unroll
        for (int v = 0; v < 8; ++v) cout[(size_t)v * NOUT] = c[v];
    } else {
#pragma unroll
        for (int v = 0; v < 8; ++v) {
            int r = m0 + half * 8 + v;
            if (r < M) cout[(size_t)v * NOUT] = c[v];
        }
    }
}

// ---------------- layer-1 aggregation (dim 64) ----------------

// bias + self-loop, vectorized float4 over N*64 elements
__global__ void k_init_agg1(const float4* __restrict__ h1,
                            const float* __restrict__ dinv,
                            const float4* __restrict__ b1v,   // 16 float4 = 64 floats
                            float4* __restrict__ agg, int n) {
    int i = blockIdx.x * blockDim.x + threadIdx.x;            // over n*16 float4s
    if (i >= n * (HID / 4)) return;
    int node = i >> 4, j = i & 15;
    float di = dinv[node];
    float s = di * di;
    float4 h = h1[i], b = b1v[j];
    float4 r;
    r.x = b.x + h.x * s; r.y = b.y + h.y * s;
    r.z = b.z + h.z * s; r.w = b.w + h.w * s;
    agg[i] = r;
}

__global__ void k_agg1_edges(const long long* __restrict__ ei,
                             const float* __restrict__ ew,
                             const float* __restrict__ dinv,
                             const float* __restrict__ h1,
                             float* __restrict__ agg, int E) {
    long long gid = (long long)blockIdx.x * blockDim.x + threadIdx.x;
    int e = (int)(gid >> 5);
    int lane = (int)(gid & 31);
    if (e >= E) return;
    int s = (int)ei[e];
    int d = (int)ei[(size_t)E + e];
    float norm = dinv[s] * ew[e] * dinv[d];
    const float* hs = h1 + (size_t)s * HID;
    float* od = agg + (size_t)d * HID;
    atomicAdd(&od[lane],      hs[lane]      * norm);
    atomicAdd(&od[lane + 32], hs[lane + 32] * norm);
}

// ---------------- layer-2 aggregation (dim 16, into d_out) ----------------

__global__ void k_init_out(const float4* __restrict__ h2,
                           const float* __restrict__ dinv,
                           const float4* __restrict__ b2v,    // 4 float4 = 16 floats
                           float4* __restrict__ out, int n) {
    int i = blockIdx.x * blockDim.x + threadIdx.x;            // over n*4 float4s
    if (i >= n * (NCLS / 4)) return;
    int node = i >> 2, j = i & 3;
    float di = dinv[node];
    float s = di * di;
    float4 h = h2[i], b = b2v[j];
    float4 r;
    r.x = b.x + h.x * s; r.y = b.y + h.y * s;
    r.z = b.z + h.z * s; r.w = b.w + h.w * s;
    out[i] = r;
}

__global__ void k_agg2_edges(const long long* __restrict__ ei,
                             const float* __restrict__ ew,
                             const float* __restrict__ dinv,
                             const float* __restrict__ h2,
                             float* __restrict__ out, int E) {
    long long gid = (long long)blockIdx.x * blockDim.x + threadIdx.x;
    int e = (int)(gid >> 4);
    int j = (int)(gid & 15);
    if (e >= E) return;
    int s = (int)ei[e];
    int d = (int)ei[(size_t)E + e];
    float norm = dinv[s] * ew[e] * dinv[d];
    atomicAdd(&out[(size_t)d * NCLS + j], h2[(size_t)s * NCLS + j] * norm);
}

// ---------------- in-place log-softmax over 16 classes ----------------

__global__ void k_logsoftmax(float* __restrict__ out, int n) {
    int i = blockIdx.x * blockDim.x + threadIdx.x;
    if (i >= n) return;
    float* r = out + (size_t)i * NCLS;
    float v[NCLS];
    float m = -3.4e38f;
#pragma unroll
    for (int j = 0; j < NCLS; ++j) { v[j] = r[j]; m = fmaxf(m, v[j]); }
    float s = 0.0f;
#pragma unroll
    for (int j = 0; j < NCLS; ++j) s += expf(v[j] - m);
    float lse = m + logf(s);
#pragma unroll
    for (int j = 0; j < NCLS; ++j) r[j] = v[j] - lse;
}

// ---------------- launch ----------------

extern "C" void kernel_launch(void* const* d_in, const int* in_sizes, int n_in,
                              void* d_out, int out_size, void* d_ws, size_t ws_size,
                              hipStream_t stream) {
    const float*     x  = (const float*)d_in[0];
    const long long* ei = (const long long*)d_in[1];  // int64 [2,E] row-major
    const float*     ew = (const float*)d_in[2];
    const float*     W1 = (const float*)d_in[3];
    const float*     b1 = (const float*)d_in[4];
    const float*     W2 = (const float*)d_in[5];
    const float*     b2 = (const float*)d_in[6];
    float* out = (float*)d_out;

    const int N = in_sizes[0] / FIN;   // 100000
    const int E = in_sizes[2];         // 1600000

    // workspace: dinv[N] | h1[N*64] | agg1[N*64]   (h2 reuses h1's buffer)
    float* dinv = (float*)d_ws;
    float* h1   = dinv + N;
    float* agg1 = h1 + (size_t)N * HID;
    float* h2   = h1;

    const int B = 256;
    const int mtiles = (N + 15) / 16;

    // degrees -> dinv
    k_init_deg<<<(N + B - 1) / B, B, 0, stream>>>(dinv, N);
    k_deg_accum<<<(E + B - 1) / B, B, 0, stream>>>(ei, ew, dinv, E);
    k_dinv<<<(N + B - 1) / B, B, 0, stream>>>(dinv, N);

    // layer 1: h1 = x @ W1  (WMMA f32)
    {
        long long waves = (long long)mtiles * (HID / 16);
        long long thr = waves * 32;
        gemm_wmma_f32<FIN, HID, false><<<(int)((thr + B - 1) / B), B, 0, stream>>>(x, W1, h1, N);
    }
    // layer 1 aggregation
    k_init_agg1<<<(int)(((long long)N * (HID / 4) + B - 1) / B), B, 0, stream>>>(
        (const float4*)h1, dinv, (const float4*)b1, (float4*)agg1, N);
    k_agg1_edges<<<(int)(((long long)E * 32 + B - 1) / B), B, 0, stream>>>(ei, ew, dinv, h1, agg1, E);

    // layer 2: h2 = relu(agg1) @ W2  (WMMA f32, ReLU fused into A load)
    {
        long long waves = (long long)mtiles * (NCLS / 16);
        long long thr = waves * 32;
        gemm_wmma_f32<HID, NCLS, true><<<(int)((thr + B - 1) / B), B, 0, stream>>>(agg1, W2, h2, N);
    }
    // layer 2 aggregation directly into d_out
    k_init_out<<<(int)(((long long)N * (NCLS / 4) + B - 1) / B), B, 0, stream>>>(
        (const float4*)h2, dinv, (const float4*)b2, (float4*)out, N);
    k_agg2_edges<<<(int)(((long long)E * NCLS + B - 1) / B), B, 0, stream>>>(ei, ew, dinv, h2, out, E);

    // log-softmax in place
    k_logsoftmax<<<(N + B - 1) / B, B, 0, stream>>>(out, N);
}